// VectorQuantizer_16226386444646
// MI455X (gfx1250) — compile-verified
//
#include <hip/hip_runtime.h>
#include <hip/hip_bf16.h>

typedef __attribute__((ext_vector_type(2))) float v2f;
typedef __attribute__((ext_vector_type(8))) float v8f;

#define N_E        8192
#define EDIM       64
#define NROWS      16384          // 16*32*32
#define MBLK       256            // rows per workgroup (8 waves * 32)
#define NPART      4              // codebook split across workgroups
#define NPER       (N_E / NPART)  // 2048 codes per partition
#define NSTAGE     64             // codebook rows staged in LDS per step
#define LDS_STRIDE 66             // 64 + 2 pad floats -> conflict-free column reads

// ---------------------------------------------------------------------------
// Kernel 1: per-code squared norms  ||e_n||^2
// ---------------------------------------------------------------------------
__global__ __launch_bounds__(256)
void vq_enorm_kernel(const float* __restrict__ emb, float* __restrict__ enorm) {
    int n = blockIdx.x * blockDim.x + threadIdx.x;
    const float4* e = (const float4*)(emb + (size_t)n * EDIM);
    float s = 0.f;
#pragma unroll
    for (int i = 0; i < EDIM / 4; ++i) {
        float4 v = e[i];
        s += v.x * v.x + v.y * v.y + v.z * v.z + v.w * v.w;
    }
    enorm[n] = s;
}

// ---------------------------------------------------------------------------
// Kernel 2: WMMA distance + running argmin.
// score(m,n) = ||e_n||^2 - 2 * z_m . e_n    (||z_m||^2 constant per row)
// A fragments pre-scaled by -2 so the f32 WMMA accumulates -2*z.e directly.
// Each wave owns TWO 16-row M tiles -> every B fragment feeds 2 WMMAs
// (halves LDS bandwidth per WMMA, two independent accumulator chains).
// ---------------------------------------------------------------------------
__global__ __launch_bounds__(256)
void vq_argmin_kernel(const float* __restrict__ z,
                      const float* __restrict__ emb,
                      const float* __restrict__ enorm,
                      float* __restrict__ pval,
                      int*   __restrict__ pidx) {
    __shared__ float lds[NSTAGE * LDS_STRIDE + NSTAGE];
    float* lds_en = &lds[NSTAGE * LDS_STRIDE];

    const int tid  = threadIdx.x;
    const int lane = tid & 31;
    const int wave = tid >> 5;
    const int l16  = lane & 15;   // row (A) / column (B/C) within a 16x16 tile
    const int g    = lane >> 4;   // lane-group: selects K pair / C row half

    const int mb         = blockIdx.x % (NROWS / MBLK);
    const int part       = blockIdx.x / (NROWS / MBLK);
    const int nbase_part = part * NPER;
    const int mbase      = mb * MBLK + wave * 32;   // 32 rows per wave

    // A fragments (ISA 16x4 f32 A layout): row M = l16,
    // VGPR0 -> K = 4*kf + 2*g, VGPR1 -> K+1. Pre-scaled by -2.
    v2f a0[16], a1[16];
    {
        const float* zrow0 = z + (size_t)(mbase + l16) * EDIM + 2 * g;
        const float* zrow1 = zrow0 + (size_t)16 * EDIM;
#pragma unroll
        for (int kf = 0; kf < 16; ++kf) {
            v2f t0 = *(const v2f*)(zrow0 + 4 * kf);
            v2f t1 = *(const v2f*)(zrow1 + 4 * kf);
            a0[kf] = t0 * (-2.0f);
            a1[kf] = t1 * (-2.0f);
        }
    }

    float minv0[8], minv1[8];
    int   mini0[8], mini1[8];
#pragma unroll
    for (int i = 0; i < 8; ++i) {
        minv0[i] = __builtin_inff(); mini0[i] = 0;
        minv1[i] = __builtin_inff(); mini1[i] = 0;
    }

    const int r  = tid >> 2;          // staging: 4 threads per codebook row
    const int cq = (tid & 3) * 16;    // 16-float quarter of the row

    for (int stage = 0; stage < NPER / NSTAGE; ++stage) {
        // Cooperative staging: 64 codebook rows x 64 floats, padded stride.
        {
            const float4* src =
                (const float4*)(emb + (size_t)(nbase_part + stage * NSTAGE + r) * EDIM + cq);
            float* dst = &lds[r * LDS_STRIDE + cq];
#pragma unroll
            for (int i = 0; i < 4; ++i) {
                float4 v = src[i];
                dst[4 * i + 0] = v.x; dst[4 * i + 1] = v.y;
                dst[4 * i + 2] = v.z; dst[4 * i + 3] = v.w;
            }
            if (tid < NSTAGE)
                lds_en[tid] = enorm[nbase_part + stage * NSTAGE + tid];
        }
        __syncthreads();

        // Prefetch next stage's tile into cache while this stage computes.
        if (stage + 1 < NPER / NSTAGE) {
            const float* nsrc =
                emb + (size_t)(nbase_part + (stage + 1) * NSTAGE + r) * EDIM + cq;
            __builtin_prefetch(nsrc, 0, 1);   // -> global_prefetch_b8
        }

#pragma unroll
        for (int sub = 0; sub < NSTAGE / 16; ++sub) {
            // B fragment (4x16 f32): column N = l16, VGPR0 -> K = 4*kf + 2*g,
            // VGPR1 -> K+1 (mirrors the documented A layout).
            v8f c0 = {};
            v8f c1 = {};
            const float* brow = &lds[(sub * 16 + l16) * LDS_STRIDE + 2 * g];
#pragma unroll
            for (int kf = 0; kf < 16; ++kf) {
                v2f b = *(const v2f*)(brow + 4 * kf);
                c0 = __builtin_amdgcn_wmma_f32_16x16x4_f32(
                         false, a0[kf], false, b, (short)0, c0, false, false);
                c1 = __builtin_amdgcn_wmma_f32_16x16x4_f32(
                         false, a1[kf], false, b, (short)0, c1, false, false);
            }
            const float en   = lds_en[sub * 16 + l16];
            const int   nidx = nbase_part + stage * NSTAGE + sub * 16 + l16;
#pragma unroll
            for (int i = 0; i < 8; ++i) {
                float s0 = c0[i] + en;                 // ||e||^2 - 2 z.e
                float s1 = c1[i] + en;
                if (s0 < minv0[i]) { minv0[i] = s0; mini0[i] = nidx; }
                if (s1 < minv1[i]) { minv1[i] = s1; mini1[i] = nidx; }
            }
        }
        __syncthreads();
    }

    // Reduce across the 16 columns (lanes 0-15 and 16-31 independently;
    // xor offsets <= 8 never cross the half-wave boundary).
#pragma unroll
    for (int t = 0; t < 2; ++t) {
#pragma unroll
        for (int v = 0; v < 8; ++v) {
            float mv = t ? minv1[v] : minv0[v];
            int   mi = t ? mini1[v] : mini0[v];
#pragma unroll
            for (int off = 8; off >= 1; off >>= 1) {
                float ov = __shfl_xor(mv, off, 32);
                int   oi = __shfl_xor(mi, off, 32);
                if (ov < mv || (ov == mv && oi < mi)) { mv = ov; mi = oi; }
            }
            if (l16 == 0) {
                // C layout: VGPR v -> row v (lanes 0-15) / row v+8 (lanes 16-31)
                int m = mbase + t * 16 + v + g * 8;
                pval[part * NROWS + m] = mv;
                pidx[part * NROWS + m] = mi;
            }
        }
    }
}

// ---------------------------------------------------------------------------
// Kernel 3: combine the NPART partitions, gather z_q, write indices.
// Partitions are scanned in ascending-index order with strict <, matching
// argmin's first-occurrence tie-break.
// ---------------------------------------------------------------------------
__global__ __launch_bounds__(256)
void vq_finalize_kernel(const float* __restrict__ pval,
                        const int*   __restrict__ pidx,
                        const float* __restrict__ emb,
                        float*       __restrict__ out) {
    int gid = blockIdx.x * blockDim.x + threadIdx.x;   // NROWS*EDIM threads
    int m   = gid >> 6;
    int col = gid & 63;
    float bv = pval[m];
    int   bi = pidx[m];
#pragma unroll
    for (int p = 1; p < NPART; ++p) {
        float v = pval[p * NROWS + m];
        int   i = pidx[p * NROWS + m];
        if (v < bv) { bv = v; bi = i; }
    }
    out[gid] = emb[(size_t)bi * EDIM + col];           // z_q == embeddings[idx]
    if (col == 0)
        out[(size_t)NROWS * EDIM + m] = (float)bi;     // indices, f32 convention
}

// ---------------------------------------------------------------------------
extern "C" void kernel_launch(void* const* d_in, const int* in_sizes, int n_in,
                              void* d_out, int out_size, void* d_ws, size_t ws_size,
                              hipStream_t stream) {
    const float* z   = (const float*)d_in[0];   // [16,32,32,64] f32
    const float* emb = (const float*)d_in[1];   // [8192,64] f32
    float* out = (float*)d_out;                 // z_q (NROWS*EDIM) ++ idx (NROWS)

    // Workspace: enorm(8192 f32) | pval(NPART*16384 f32) | pidx(NPART*16384 i32)
    float* enorm = (float*)d_ws;
    float* pval  = enorm + N_E;
    int*   pidx  = (int*)(pval + NPART * NROWS);

    vq_enorm_kernel<<<N_E / 256, 256, 0, stream>>>(emb, enorm);
    vq_argmin_kernel<<<(NROWS / MBLK) * NPART, 256, 0, stream>>>(z, emb, enorm, pval, pidx);
    vq_finalize_kernel<<<(NROWS * EDIM) / 256, 256, 0, stream>>>(pval, pidx, emb, out);
}